// MotionCompensation_36584531427946
// MI455X (gfx1250) — compile-verified
//
#include <hip/hip_runtime.h>

// Bilinear warp (motion compensation), memory-bound kernel for MI455X (gfx1250).
// x: [B,H,W,3] float32 interleaved (frame, flow_x, flow_y); out: [B,H,W] float32.

typedef float float4v  __attribute__((ext_vector_type(4)));              // 16B aligned
typedef float float4u  __attribute__((ext_vector_type(4), aligned(4)));  // dword aligned

#define MC_B 16
#define MC_H 720
#define MC_W 1280

__global__ __launch_bounds__(320) void MotionCompensation_kernel(
    const float* __restrict__ x, float* __restrict__ out) {
  constexpr int B = MC_B, H = MC_H, W = MC_W;
  constexpr int rowElems = W * 3;            // floats per image row (3840)
  constexpr int rowBytes = rowElems * 4;     // 15360 B, fits 24-bit IOFFSET

  const int y   = blockIdx.x;                // image row
  const int b   = blockIdx.y;                // batch
  const int row = b * H + y;                 // global row id
  const int tid = threadIdx.x;               // 0..319, 4 pixels each (320*4 == W)

  const float* rowPtr = x + (size_t)row * rowElems;            // this row, interleaved
  const float* inp0   = x + (size_t)b * H * (size_t)rowElems;  // batch base (uniform)

  // Coalesced stream: 48 contiguous bytes per thread, 16B-aligned -> 3x global_load_b128.
  const float4v* p = (const float4v*)(rowPtr + (size_t)tid * 12);
  const float4v r0 = p[0];
  const float4v r1 = p[1];
  const float4v r2 = p[2];

  // gfx1250 prefetch path: warm next row's lines (taps with dy>0 read them).
  if (row + 1 < B * H) {
    __builtin_prefetch((const void*)(rowPtr + rowElems + (size_t)tid * 12), 0, 3);
  }

  // De-interleave the 12 streamed floats into per-pixel (flow_x, flow_y).
  const float wx[4] = {r0.y, r1.x, r1.w, r2.z};
  const float wy[4] = {r0.z, r1.y, r2.x, r2.w};

  const int   x0 = tid * 4;
  const float fy_base = (float)y;

  float res[4];
#pragma unroll
  for (int j = 0; j < 4; ++j) {
    const float fx = (float)(x0 + j) + wx[j];
    const float fy = fy_base + wy[j];

    // Truncate toward zero (matches jnp float->int32 cast), then clamp.
    int cx = (int)fx;  cx = min(max(cx, 0), W - 2);
    int cy = (int)fy;  cy = min(max(cy, 0), H - 2);

    const float dx  = fx - (float)cx;
    const float dy  = fy - (float)cy;
    const float ndx = 1.0f - dx;
    const float ndy = 1.0f - dy;

    // Taps (cy,cx) and (cy,cx+1) are dwords 0 and 3 of one 16B span (channel
    // stride = 3 floats): one dword-aligned b128 load covers both. Row cy+1 is
    // the same VGPR address + immediate offset 15360 (second b128).
    // 32-bit byte offset (batch input is 11 MB) -> saddr + voffset addressing.
    const unsigned byteOff =
        ((unsigned)cy * (unsigned)rowElems + (unsigned)cx * 3u) * 4u;
    const float4u ta = *(const float4u*)((const char*)inp0 + byteOff);
    const float4u tb = *(const float4u*)((const char*)inp0 + byteOff + rowBytes);

    const float t00 = ta.x;   // (cy,   cx  )
    const float t01 = ta.w;   // (cy,   cx+1)
    const float t10 = tb.x;   // (cy+1, cx  )
    const float t11 = tb.w;   // (cy+1, cx+1)

    res[j] = fmaf(t00, ndx * ndy,
             fmaf(t01, dx  * ndy,
             fmaf(t11, dx  * dy,
                  t10 * (ndx * dy))));
  }

  // Write-once output: non-temporal b128 store keeps L2 (192MB) for the input
  // lines the gathers depend on.
  float4v o;
  o.x = res[0]; o.y = res[1]; o.z = res[2]; o.w = res[3];
  float4v* op = (float4v*)(out + (size_t)row * W + x0);
  __builtin_nontemporal_store(o, op);
}

extern "C" void kernel_launch(void* const* d_in, const int* in_sizes, int n_in,
                              void* d_out, int out_size, void* d_ws, size_t ws_size,
                              hipStream_t stream) {
  (void)in_sizes; (void)n_in; (void)d_ws; (void)ws_size; (void)out_size;
  const float* x = (const float*)d_in[0];
  float* out = (float*)d_out;
  dim3 grid(MC_H, MC_B);    // y-rows x batches: no integer division in-kernel
  dim3 block(320);          // 10 wave32s, 4 px/thread
  MotionCompensation_kernel<<<grid, block, 0, stream>>>(x, out);
}